// SRPooler_77249281786347
// MI455X (gfx1250) — compile-verified
//
#include <hip/hip_runtime.h>
#include <cstdint>

// FPN RoIAlign for MI455X (gfx1250).
// One block per (ROI, 128-channel half). 16 slabs of 8 channels are staged
// through a double-buffered LDS pipeline using Tensor Data Mover descriptors:
//   barrier -> wave0 issues TDM(slab s+1) into buf[(s+1)&1]
//           -> s_wait_tensorcnt(1)  (slab s complete, s+1 still streaming)
//   barrier -> all waves do bilinear sampling of slab s from LDS.
// Gather-bound op: TDM replaces per-lane scattered global loads with strided
// 3D tile DMAs and overlaps them with the LDS compute phase.

typedef uint32_t u32;
typedef uint64_t u64;
typedef u32 __attribute__((ext_vector_type(4))) v4u;
typedef int __attribute__((ext_vector_type(4))) v4i;
typedef int __attribute__((ext_vector_type(8))) v8i;

#define POOL   7
#define CB     8            // channels per slab (per TDM)
#define SLABS  16           // slabs per block -> 128 channels per block
#define PMAX   32           // max patch dim (box*scale <= ~30 for this mapping)
#define BUFELT (CB * PMAX * PMAX)   // 8192 floats = 32 KB per buffer

__global__ __launch_bounds__(256) void
roi_align_fpn_tdm(const float* __restrict__ x0, const float* __restrict__ x1,
                  const float* __restrict__ x2, const float* __restrict__ x3,
                  const float* __restrict__ rois, float* __restrict__ out,
                  int C)
{
    extern __shared__ float smem[];   // 2 * BUFELT floats (64 KB)

    const int n      = blockIdx.y;            // ROI index
    const int c_base = blockIdx.x * (SLABS * CB);   // 0 or 128

    // ---- per-ROI params (uniform across the block; computed once) ----
    const float rb  = rois[n * 5 + 0];
    const float rx1 = rois[n * 5 + 1];
    const float ry1 = rois[n * 5 + 2];
    const float rx2 = rois[n * 5 + 3];
    const float ry2 = rois[n * 5 + 4];
    const int   b   = (int)rb;

    // level mapping: floor(4 + log2(sqrt(w*h)/224 + 1e-6)) clipped to [2,5]
    const float sdim = sqrtf((rx2 - rx1) * (ry2 - ry1));
    float tgt = floorf(4.0f + log2f(sdim * (1.0f / 224.0f) + 1e-6f));
    tgt = fminf(fmaxf(tgt, 2.0f), 5.0f);
    const int lvl = (int)tgt - 2;

    const float* feat;
    int H, W;
    float scale;
    if (lvl == 0)      { feat = x0; H = 200; W = 336; scale = 0.25f;    }
    else if (lvl == 1) { feat = x1; H = 100; W = 168; scale = 0.125f;   }
    else if (lvl == 2) { feat = x2; H = 50;  W = 84;  scale = 0.0625f;  }
    else               { feat = x3; H = 25;  W = 42;  scale = 0.03125f; }

    const float x1c = rx1 * scale, y1c = ry1 * scale;
    const float x2c = rx2 * scale, y2c = ry2 * scale;
    const float rw = fmaxf(x2c - x1c, 1.0f);
    const float rh = fmaxf(y2c - y1c, 1.0f);
    const float bw = rw * (1.0f / POOL);
    const float bh = rh * (1.0f / POOL);

    // patch origin & extent covering all clamped bilinear taps
    // (sample range along an axis: [lo + 0.25*b, lo + 6.75*b])
    float mnx = fminf(fmaxf(x1c + 0.25f * bw, 0.0f), (float)(W - 1));
    float mny = fminf(fmaxf(y1c + 0.25f * bh, 0.0f), (float)(H - 1));
    float mxx = fminf(fmaxf(x1c + 6.75f * bw, 0.0f), (float)(W - 1));
    float mxy = fminf(fmaxf(y1c + 6.75f * bh, 0.0f), (float)(H - 1));
    const int px0 = (int)floorf(mnx);
    const int py0 = (int)floorf(mny);
    int pxe = min((int)floorf(mxx) + 1, W - 1);
    int pye = min((int)floorf(mxy) + 1, H - 1);
    const int pw_p = min(pxe - px0 + 1, PMAX);
    const int ph_p = min(pye - py0 + 1, PMAX);

    // ---- TDM descriptor pieces shared by all slabs of this ROI ----
    const u64 groi = (u64)(uintptr_t)feat +
        4ull * ((u64)b * (u64)C * (u64)(H * W) +
                (u64)py0 * (u64)W + (u64)px0);
    const u32 lds_base = (u32)(uintptr_t)smem;   // low 32 bits = LDS byte addr

    const u32 td0 = (u32)(W - px0);
    const u32 td1 = (u32)(H - py0);
    const u32 st1 = (u32)(H * W);
    v8i g1;
    g1[0] = (int)0x00020000u;                               // data_size=2 -> 4B
    g1[1] = (int)((td0 & 0xffffu) << 16);                   // tensor_dim0[15:0]
    g1[2] = (int)((td0 >> 16) | ((td1 & 0xffffu) << 16));   // dim0 hi | dim1 lo
    g1[3] = (int)((td1 >> 16) | ((u32)pw_p << 16));         // dim1 hi | tile_dim0
    g1[4] = (int)((u32)ph_p | ((u32)CB << 16));             // tile_dim1 | tile_dim2
    g1[5] = (int)(u32)W;                                    // dim0_stride[31:0]
    g1[6] = (int)((st1 & 0xffffu) << 16);                   // dim0_str hi=0 | dim1_str lo
    g1[7] = (int)(st1 >> 16);                               // dim1_stride[47:16]
    const v4i g2 = { 256, 0, 0, 0 };   // tensor_dim2 (>= CB); dims 3/4 unused
    const v4i g3 = { 0, 0, 0, 0 };
    const v8i gx = { 0, 0, 0, 0, 0, 0, 0, 0 };  // clang-23 6-arg form: ext group

    auto issue_slab = [&](int slab) {
        const u64 gaddr = groi + 4ull * (u64)(c_base + slab * CB) * (u64)(H * W);
        v4u g0;
        g0.x = 1u;                                          // count=1
        g0.y = lds_base + (u32)((slab & 1) * (BUFELT * 4)); // lds_addr
        g0.z = (u32)(gaddr & 0xffffffffull);
        g0.w = (u32)((gaddr >> 32) & 0x01ffffffull) | 0x80000000u;  // type=2
        __builtin_amdgcn_tensor_load_to_lds(g0, g1, g2, g3, gx, 0);
    };

    const bool issuer = (threadIdx.x < 32);   // wave32: first wave runs the DMA
    if (issuer) issue_slab(0);

    const int   stride_c = pw_p * ph_p;
    const float fW1 = (float)(W - 1), fH1 = (float)(H - 1);

    for (int s = 0; s < SLABS; ++s) {
        __syncthreads();   // everyone done reading buf[(s+1)&1] (slab s-1)
        if (issuer) {
            if (s + 1 < SLABS) {
                issue_slab(s + 1);                          // prefetch next slab
                __builtin_amdgcn_s_wait_tensorcnt(1);       // slab s landed
            } else {
                __builtin_amdgcn_s_wait_tensorcnt(0);
            }
        }
        __syncthreads();   // slab s visible to all waves

        const float* buf = smem + (s & 1) * BUFELT;
        const int    c0  = c_base + s * CB;

        for (int o = threadIdx.x; o < CB * POOL * POOL; o += blockDim.x) {
            const int c   = o / (POOL * POOL);    // constant divisor -> mulhi
            const int bin = o % (POOL * POOL);
            const int py  = bin / POOL;
            const int px  = bin % POOL;
            const float* pc = buf + c * stride_c;

            float acc = 0.0f;
            #pragma unroll
            for (int si = 0; si < 2; ++si) {
                const float sy = y1c + (float)py * bh + ((float)si + 0.5f) * bh * 0.5f;
                const float my = (sy > -1.0f && sy < (float)H) ? 1.0f : 0.0f;
                const float yc = fminf(fmaxf(sy, 0.0f), fH1);
                const int   yi0 = (int)floorf(yc);
                const int   yi1 = min(yi0 + 1, H - 1);
                const float ly = yc - (float)yi0, hy = 1.0f - ly;
                const int ty0 = min(max(yi0 - py0, 0), ph_p - 1);
                const int ty1 = min(max(yi1 - py0, 0), ph_p - 1);
                #pragma unroll
                for (int sj = 0; sj < 2; ++sj) {
                    const float sx = x1c + (float)px * bw + ((float)sj + 0.5f) * bw * 0.5f;
                    const float mx = (sx > -1.0f && sx < (float)W) ? 1.0f : 0.0f;
                    const float xc = fminf(fmaxf(sx, 0.0f), fW1);
                    const int   xi0 = (int)floorf(xc);
                    const int   xi1 = min(xi0 + 1, W - 1);
                    const float lx = xc - (float)xi0, hx = 1.0f - lx;
                    const int tx0 = min(max(xi0 - px0, 0), pw_p - 1);
                    const int tx1 = min(max(xi1 - px0, 0), pw_p - 1);

                    const float v00 = pc[ty0 * pw_p + tx0];
                    const float v01 = pc[ty0 * pw_p + tx1];
                    const float v10 = pc[ty1 * pw_p + tx0];
                    const float v11 = pc[ty1 * pw_p + tx1];
                    const float v = hy * (hx * v00 + lx * v01) + ly * (hx * v10 + lx * v11);
                    acc += v * (my * mx);
                }
            }
            out[((u64)n * (u64)C + (u64)(c0 + c)) * (POOL * POOL) + bin] = acc * 0.25f;
        }
    }
}

extern "C" void kernel_launch(void* const* d_in, const int* in_sizes, int n_in,
                              void* d_out, int out_size, void* d_ws, size_t ws_size,
                              hipStream_t stream) {
    const float* x0   = (const float*)d_in[0];
    const float* x1   = (const float*)d_in[1];
    const float* x2   = (const float*)d_in[2];
    const float* x3   = (const float*)d_in[3];
    const float* rois = (const float*)d_in[4];
    float* out = (float*)d_out;

    const int N = in_sizes[4] / 5;   // 1024 ROIs
    const int C = 256;

    const dim3 grid(C / (SLABS * CB), N);   // (2, 1024): channel half x ROI
    const dim3 block(256);
    const size_t shmem = 2u * BUFELT * sizeof(float);  // 64 KB double buffer
    roi_align_fpn_tdm<<<grid, block, shmem, stream>>>(x0, x1, x2, x3, rois, out, C);
}